// Net_39041252720977
// MI455X (gfx1250) — compile-verified
//
#include <hip/hip_runtime.h>
#include <math.h>

typedef __attribute__((ext_vector_type(2))) float v2f;
typedef __attribute__((ext_vector_type(4))) float v4f;
typedef __attribute__((ext_vector_type(8))) float v8f;

#define NN   10000
#define EE   320000
#define KDIM 256
#define DOUT 40
#define DOUTP 48   // DOUT padded to tile multiple

struct __align__(8) Edge { int s; float w; };   // packed CSR edge record (one b64 load)

// ---------------------------------------------------------------- CSR build
__global__ void count_kernel(const int* __restrict__ dst, int* __restrict__ counts) {
    int i = blockIdx.x * blockDim.x + threadIdx.x;
    if (i < EE) atomicAdd(&counts[dst[i]], 1);
}

__global__ void dinv_kernel(const int* __restrict__ counts, float* __restrict__ dinv) {
    int i = blockIdx.x * blockDim.x + threadIdx.x;
    if (i < NN) dinv[i] = rsqrtf((float)counts[i] + 1.0f);  // +1 for self loop
}

// single-block two-level exclusive scan of counts -> row_ptr (and cursor copy)
__global__ void scan_kernel(const int* __restrict__ counts,
                            int* __restrict__ row_ptr, int* __restrict__ cursor) {
    __shared__ int partial[256];
    const int CH = (NN + 255) / 256;      // 40 elements per thread
    int t = threadIdx.x;
    int base = t * CH;
    int s = 0;
    for (int j = 0; j < CH; ++j) { int idx = base + j; if (idx < NN) s += counts[idx]; }
    partial[t] = s;
    __syncthreads();
    if (t == 0) {
        int run = 0;
        for (int i = 0; i < 256; ++i) { int v = partial[i]; partial[i] = run; run += v; }
    }
    __syncthreads();
    int run = partial[t];
    for (int j = 0; j < CH; ++j) {
        int idx = base + j;
        if (idx < NN) { row_ptr[idx] = run; cursor[idx] = run; run += counts[idx]; }
    }
    if (t == 255) row_ptr[NN] = run;       // == EE
}

__global__ void fill_kernel(const int* __restrict__ src, const int* __restrict__ dst,
                            const float* __restrict__ dinv, int* __restrict__ cursor,
                            Edge* __restrict__ edges) {
    int i = blockIdx.x * blockDim.x + threadIdx.x;
    if (i >= EE) return;
    int d  = dst[i];
    int sc = src[i];
    int p  = atomicAdd(&cursor[d], 1);
    Edge e; e.s = sc; e.w = dinv[sc] * dinv[d];
    edges[p] = e;
}

// ---------------------------------------------------------------- weight transpose (+pad)
// out[c*KDIM + k] = (c < cols) ? in[k*cols + c] : 0   for c in [0, colsPad)
__global__ void wtranspose_kernel(const float* __restrict__ in, float* __restrict__ out,
                                  int cols, int colsPad) {
    int idx = blockIdx.x * blockDim.x + threadIdx.x;
    if (idx >= colsPad * KDIM) return;
    int c = idx / KDIM, k = idx - c * KDIM;
    out[idx] = (c < cols) ? in[k * cols + c] : 0.0f;
}

// ---------------------------------------------------------------- fp32 WMMA GEMM
// out[M x ncols] = A[M x 256] * W[256 x ncols]; W supplied TRANSPOSED+PADDED as
// Wt[colsPad x 256]. One 16x16 tile per wave, 8 waves/block share a 16x256 A
// panel in LDS (stride 260: bank-conflict-free, 8B aligned for b64 loads).
// Inner K-loop is branch-free: ds_load_2addr_b64 + global_load_b64 + wmma.
__global__ __launch_bounds__(256) void gemm_wmma_f32(const float* __restrict__ A,
                                                     const float* __restrict__ Wt,
                                                     float* __restrict__ out,
                                                     int ncols, int ntiles) {
    __shared__ float ldsA[16 * 260];
    const int m0 = blockIdx.x * 16;
    const int t  = threadIdx.x;

    for (int idx = t; idx < 16 * 256; idx += 256) {
        int r = idx >> 8, c = idx & 255;
        ldsA[r * 260 + c] = A[(m0 + r) * KDIM + c];
    }
    __syncthreads();

    const int wave = t >> 5;
    const int lane = t & 31;
    const int nt   = blockIdx.y * 8 + wave;
    if (nt >= ntiles) return;              // wave-uniform exit (EXEC stays full)

    const int l15  = lane & 15;            // A row (m) within tile / B column (n)
    const int half = lane >> 4;            // k-half selector
    const int col  = nt * 16 + l15;        // padded column index (always valid in Wt)

    const float* __restrict__ arow = &ldsA[l15 * 260];
    const float* __restrict__ wcol = &Wt[col * KDIM];

    v8f acc = {};
#pragma unroll 8
    for (int kk = 0; kk < KDIM; kk += 4) {
        int ka = kk + 2 * half;
        v2f a = *(const v2f*)(arow + ka);   // {A[m][ka], A[m][ka+1]}
        v2f b = *(const v2f*)(wcol + ka);   // {W[ka][col], W[ka+1][col]}
        acc = __builtin_amdgcn_wmma_f32_16x16x4_f32(
                  false, a, false, b, (short)0, acc, false, false);
    }

    if (col < ncols) {
#pragma unroll
        for (int r = 0; r < 8; ++r) {
            int row = m0 + r + 8 * half;   // C layout: VGPR r -> M=r (lanes0-15) / M=r+8
            out[row * ncols + col] = acc[r];
        }
    }
}

// ---------------------------------------------------------------- CSR aggregation
// 256-wide layers: one block of 64 threads per dst node, each thread owns a
// float4 feature slice -> per edge: 1 broadcast b64 (edge rec) + 64 coalesced
// global_load_b128 (1KB row gather, L2-resident) + 4 FMAs.
__global__ __launch_bounds__(64) void aggregate256_kernel(const float* __restrict__ proj,
                                                          const float* __restrict__ dinv,
                                                          const int* __restrict__ row_ptr,
                                                          const Edge* __restrict__ edges,
                                                          const float* __restrict__ bias,
                                                          float* __restrict__ out) {
    const int i = blockIdx.x;
    const int t = threadIdx.x;             // 0..63, feature quad t
    const v4f* __restrict__ proj4 = (const v4f*)proj;
    int e0 = row_ptr[i], e1 = row_ptr[i + 1];
    v4f acc = {};
    for (int e = e0; e < e1; ++e) {
        Edge ed = edges[e];                // single b64 broadcast load
        acc += ed.w * proj4[ed.s * (KDIM / 4) + t];
    }
    float dv = dinv[i];
    acc += (dv * dv) * proj4[i * (KDIM / 4) + t];   // self-loop term
    acc += ((const v4f*)bias)[t];
    acc = __builtin_elementwise_max(acc, (v4f)0.0f); // relu
    ((v4f*)out)[i * (KDIM / 4) + t] = acc;
}

// 40-wide last layer: scalar, 64 threads/node (40 active), no relu
__global__ __launch_bounds__(64) void aggregate40_kernel(const float* __restrict__ proj,
                                                         const float* __restrict__ dinv,
                                                         const int* __restrict__ row_ptr,
                                                         const Edge* __restrict__ edges,
                                                         const float* __restrict__ bias,
                                                         float* __restrict__ out) {
    const int i = blockIdx.x;
    const int t = threadIdx.x;
    if (t >= DOUT) return;
    int e0 = row_ptr[i], e1 = row_ptr[i + 1];
    float acc = 0.0f;
    for (int e = e0; e < e1; ++e) {
        Edge ed = edges[e];
        acc += ed.w * proj[ed.s * DOUT + t];
    }
    float dv = dinv[i];
    acc += dv * dv * proj[i * DOUT + t];
    acc += bias[t];
    out[i * DOUT + t] = acc;
}

// ---------------------------------------------------------------- log_softmax (40-wide)
__global__ void logsoftmax_kernel(const float* __restrict__ in, float* __restrict__ out) {
    int i = blockIdx.x * blockDim.x + threadIdx.x;
    if (i >= NN) return;
    const float* r = in + i * DOUT;
    float m = -3.402823466e38f;
    for (int j = 0; j < DOUT; ++j) m = fmaxf(m, r[j]);
    float s = 0.0f;
    for (int j = 0; j < DOUT; ++j) s += expf(r[j] - m);
    float ls = logf(s) + m;
    for (int j = 0; j < DOUT; ++j) out[i * DOUT + j] = r[j] - ls;
}

// ---------------------------------------------------------------- launcher
static inline size_t align256(size_t x) { return (x + 255u) & ~(size_t)255u; }

extern "C" void kernel_launch(void* const* d_in, const int* in_sizes, int n_in,
                              void* d_out, int out_size, void* d_ws, size_t ws_size,
                              hipStream_t stream) {
    (void)in_sizes; (void)n_in; (void)out_size; (void)ws_size;

    const float* x  = (const float*)d_in[0];
    const int*   ei = (const int*)d_in[1];
    const float* W1 = (const float*)d_in[2];
    const float* b1 = (const float*)d_in[3];
    const float* Wh = (const float*)d_in[4];
    const float* bh = (const float*)d_in[5];
    const float* W2 = (const float*)d_in[6];
    const float* b2 = (const float*)d_in[7];
    const int* src = ei;          // edge_index[0]
    const int* dst = ei + EE;     // edge_index[1]
    float* outp = (float*)d_out;

    // workspace carve-up
    char* ws = (char*)d_ws;
    size_t off = 0;
    int*   counts  = (int*)(ws + off);   off += align256((size_t)NN * 4);
    int*   row_ptr = (int*)(ws + off);   off += align256((size_t)(NN + 1) * 4);
    int*   cursor  = (int*)(ws + off);   off += align256((size_t)NN * 4);
    float* dinv    = (float*)(ws + off); off += align256((size_t)NN * 4);
    Edge*  edges   = (Edge*)(ws + off);  off += align256((size_t)EE * 8);
    float* Wt1     = (float*)(ws + off); off += align256((size_t)KDIM * KDIM * 4);
    float* Wth     = (float*)(ws + off); off += align256((size_t)KDIM * KDIM * 4);
    float* Wt2     = (float*)(ws + off); off += align256((size_t)DOUTP * KDIM * 4);
    float* P       = (float*)(ws + off); off += align256((size_t)NN * KDIM * 4);
    float* H1      = (float*)(ws + off); off += align256((size_t)NN * KDIM * 4);
    float* H2      = (float*)(ws + off); off += align256((size_t)NN * KDIM * 4);
    float* H3      = (float*)(ws + off); off += align256((size_t)NN * DOUT * 4);

    // --- weight transposes (L2-resident, branch-free GEMM B-side) ---
    wtranspose_kernel<<<(KDIM * KDIM + 255) / 256, 256, 0, stream>>>(W1, Wt1, KDIM, KDIM);
    wtranspose_kernel<<<(KDIM * KDIM + 255) / 256, 256, 0, stream>>>(Wh, Wth, KDIM, KDIM);
    wtranspose_kernel<<<(DOUTP * KDIM + 255) / 256, 256, 0, stream>>>(W2, Wt2, DOUT, DOUTP);

    // --- GCN normalization + CSR build ---
    hipMemsetAsync(counts, 0, (size_t)NN * 4, stream);
    count_kernel<<<(EE + 255) / 256, 256, 0, stream>>>(dst, counts);
    dinv_kernel<<<(NN + 255) / 256, 256, 0, stream>>>(counts, dinv);
    scan_kernel<<<1, 256, 0, stream>>>(counts, row_ptr, cursor);
    fill_kernel<<<(EE + 255) / 256, 256, 0, stream>>>(src, dst, dinv, cursor, edges);

    // --- layer 1: relu(conv(x, W1, b1)) ---
    gemm_wmma_f32<<<dim3(NN / 16, 2), 256, 0, stream>>>(x, Wt1, P, KDIM, KDIM / 16);
    aggregate256_kernel<<<NN, 64, 0, stream>>>(P, dinv, row_ptr, edges, b1, H1);

    // --- layer 2: relu(conv(h1, Wh, bh)) ---
    gemm_wmma_f32<<<dim3(NN / 16, 2), 256, 0, stream>>>(H1, Wth, P, KDIM, KDIM / 16);
    aggregate256_kernel<<<NN, 64, 0, stream>>>(P, dinv, row_ptr, edges, bh, H2);

    // --- layer 3: conv(h2, W2, b2), 40-wide (3 padded n-tiles, store-guarded) ---
    gemm_wmma_f32<<<dim3(NN / 16, 1), 256, 0, stream>>>(H2, Wt2, P, DOUT, 3);
    aggregate40_kernel<<<NN, 64, 0, stream>>>(P, dinv, row_ptr, edges, b2, H3);

    // --- log_softmax ---
    logsoftmax_kernel<<<(NN + 255) / 256, 256, 0, stream>>>(H3, outp);
}